// WFRSinkhornLoss_52802327937049
// MI455X (gfx1250) — compile-verified
//
#include <hip/hip_runtime.h>

typedef float v2f __attribute__((ext_vector_type(2)));
typedef float v8f __attribute__((ext_vector_type(8)));

#define IJ      256
#define PITCH   260            // 256 + 4 words: row pass conflict-free, col pass ~2-way on 8 banks
#define NITER   50
#define EPS     0.1f
#define INV_EPS 10.0f
#define LOG2E   1.4426950408889634f
#define SSCALE  (INV_EPS * LOG2E)       // u,v,C stored pre-scaled by this; K = exp2(U + V - C')
#define PCOEF   (1.0f / 1.1f)           // 1/(1+eps)
#define DXY     (1.0f / 256.0f)
#define HUGEV   1.0e30f
#define SMALLV  1.0e-10f
#define PI_2    1.5707963267948966f

__device__ __forceinline__ v8f wmma_acc(v2f a, v2f b, v8f c) {
  // D(16x16,f32) += A(16x4,f32) x B(4x16,f32)
  return __builtin_amdgcn_wmma_f32_16x16x4_f32(false, a, false, b, (short)0, c, false, false);
}
__device__ __forceinline__ float ex2(float x) { return __builtin_amdgcn_exp2f(x); }   // v_exp_f32
__device__ __forceinline__ float lg2(float x) { return __builtin_amdgcn_logf(x); }    // v_log_f32

__global__ __launch_bounds__(256, 1) void wfr_sinkhorn_kernel(const float* __restrict__ D,
                                                              float* __restrict__ out) {
  extern __shared__ float Csh[];  // 256 * PITCH f32 = 260 KB, LDS-resident pre-scaled cost matrix
  __shared__ float u[IJ], v[IJ], unew[IJ], wv[IJ], srow[IJ], scol[IJ], tred[8];

  const int tid  = threadIdx.x;
  const int bat  = blockIdx.x;
  const int lane = tid & 31;
  const int wave = tid >> 5;
  const int m    = lane & 15;          // WMMA M (or N-col in col pass)
  const int kb   = (lane >> 4) << 1;   // K sub-offset: lanes 0-15 -> {0,1}, 16-31 -> {2,3}

  // ---- load D, transform to C' = C*S = -20*log2(cos(min(2D, pi/2)) + 1e-5), store to LDS
  const float* Db = D + (size_t)bat * (IJ * IJ);
  for (int idx = tid; idx < IJ * IJ; idx += 256) {
    const int i = idx >> 8, j = idx & 255;
    const float d = Db[idx];
    Csh[i * PITCH + j] = -20.0f * lg2(__cosf(fminf(d * 2.0f, PI_2)) + 1e-5f);
  }
  u[tid] = 0.0f;   // stored in units of S*u (log2-exponent units)
  v[tid] = 0.0f;
  __syncthreads();

  const v8f vz = {0.f, 0.f, 0.f, 0.f, 0.f, 0.f, 0.f, 0.f};

  for (int it = 0; it < NITER; ++it) {
    // ---- row pass: srow_i = sum_j exp2(U_i + V_j - C'_ij) * dy   (b == 1: per-iter rescale)
    for (int gp = 0; gp < 2; ++gp) {
      const int   r0   = (wave + gp * 8) * 16;
      const float um   = u[r0 + m];
      const float* crow = &Csh[(r0 + m) * PITCH];
      v8f acc0 = vz, acc1 = vz;
      for (int j0 = 0; j0 < IJ; j0 += 8) {
        {
          const int j = j0 + kb;
          const float e0 = fminf(ex2(um + v[j]     - crow[j]),     HUGEV);
          const float e1 = fminf(ex2(um + v[j + 1] - crow[j + 1]), HUGEV);
          v2f a = {e0, e1}; v2f bw = {DXY, DXY};
          acc0 = wmma_acc(a, bw, acc0);
        }
        {
          const int j = j0 + 4 + kb;
          const float e0 = fminf(ex2(um + v[j]     - crow[j]),     HUGEV);
          const float e1 = fminf(ex2(um + v[j + 1] - crow[j + 1]), HUGEV);
          v2f a = {e0, e1}; v2f bw = {DXY, DXY};
          acc1 = wmma_acc(a, bw, acc1);
        }
      }
      if (m == 0) {  // lane 0 -> rows r0+0..7, lane 16 -> rows r0+8..15
        const int rb = r0 + ((lane >> 4) << 3);
        #pragma unroll
        for (int r = 0; r < 8; ++r) srow[rb + r] = acc0[r] + acc1[r];
      }
    }
    __syncthreads();

    // ---- a-update: log2(a) = -p*(0.1*U + log2 s); rescale U' = U + log2(a_clipped)
    {
      const float s  = srow[tid];
      const float la = -PCOEF * (EPS * u[tid] + lg2(s));
      const float a  = fminf(ex2(la), HUGEV);
      wv[tid]   = a * DXY;                 // col-pass weight a_i * dx
      unew[tid] = u[tid] + lg2(a);         // applied AFTER col pass (K uses old u)
    }
    __syncthreads();

    // ---- col pass: scol_j = sum_i exp2(U_i + V_j - C'_ij) * a_i * dx
    for (int gp = 0; gp < 2; ++gp) {
      const int   c0 = (wave + gp * 8) * 16;
      const float vm = v[c0 + m];
      v8f acc0 = vz, acc1 = vz;
      for (int i0 = 0; i0 < IJ; i0 += 8) {
        {
          const int i = i0 + kb;
          const float e0 = fminf(ex2(u[i]     + vm - Csh[i * PITCH + c0 + m]),       HUGEV);
          const float e1 = fminf(ex2(u[i + 1] + vm - Csh[(i + 1) * PITCH + c0 + m]), HUGEV);
          v2f a = {e0, e1}; v2f bw = {wv[i], wv[i + 1]};
          acc0 = wmma_acc(a, bw, acc0);
        }
        {
          const int i = i0 + 4 + kb;
          const float e0 = fminf(ex2(u[i]     + vm - Csh[i * PITCH + c0 + m]),       HUGEV);
          const float e1 = fminf(ex2(u[i + 1] + vm - Csh[(i + 1) * PITCH + c0 + m]), HUGEV);
          v2f a = {e0, e1}; v2f bw = {wv[i], wv[i + 1]};
          acc1 = wmma_acc(a, bw, acc1);
        }
      }
      if (m == 0) {
        const int cb = c0 + ((lane >> 4) << 3);
        #pragma unroll
        for (int r = 0; r < 8; ++r) scol[cb + r] = acc0[r] + acc1[r];
      }
    }
    __syncthreads();

    // ---- b-update + absorb both scalings (b back to 1 for next iteration)
    {
      const float s2 = scol[tid];
      const float lb = -PCOEF * (EPS * v[tid] + lg2(s2));
      const float bn = fminf(ex2(lb), HUGEV);
      const float vn = v[tid] + lg2(bn);
      u[tid] = unew[tid];
      v[tid] = vn;
    }
    __syncthreads();
  }

  // ---- final: row/col marginals of K(u,v) and transport = sum K*C  (C = C'/S)
  float tacc = 0.0f;
  for (int gp = 0; gp < 2; ++gp) {
    const int   r0   = (wave + gp * 8) * 16;
    const float um   = u[r0 + m];
    const float* crow = &Csh[(r0 + m) * PITCH];
    v8f acc0 = vz, acc1 = vz;
    for (int j0 = 0; j0 < IJ; j0 += 8) {
      {
        const int j = j0 + kb;
        const float c0v = crow[j], c1v = crow[j + 1];
        const float e0 = fminf(ex2(um + v[j]     - c0v), HUGEV);
        const float e1 = fminf(ex2(um + v[j + 1] - c1v), HUGEV);
        tacc += e0 * c0v + e1 * c1v;
        v2f a = {e0, e1}; v2f bw = {DXY, DXY};
        acc0 = wmma_acc(a, bw, acc0);
      }
      {
        const int j = j0 + 4 + kb;
        const float c0v = crow[j], c1v = crow[j + 1];
        const float e0 = fminf(ex2(um + v[j]     - c0v), HUGEV);
        const float e1 = fminf(ex2(um + v[j + 1] - c1v), HUGEV);
        tacc += e0 * c0v + e1 * c1v;
        v2f a = {e0, e1}; v2f bw = {DXY, DXY};
        acc1 = wmma_acc(a, bw, acc1);
      }
    }
    if (m == 0) {
      const int rb = r0 + ((lane >> 4) << 3);
      #pragma unroll
      for (int r = 0; r < 8; ++r) srow[rb + r] = acc0[r] + acc1[r];  // row_marg_i
    }
  }
  for (int gp = 0; gp < 2; ++gp) {
    const int   c0 = (wave + gp * 8) * 16;
    const float vm = v[c0 + m];
    v8f acc0 = vz, acc1 = vz;
    for (int i0 = 0; i0 < IJ; i0 += 8) {
      {
        const int i = i0 + kb;
        const float e0 = fminf(ex2(u[i]     + vm - Csh[i * PITCH + c0 + m]),       HUGEV);
        const float e1 = fminf(ex2(u[i + 1] + vm - Csh[(i + 1) * PITCH + c0 + m]), HUGEV);
        v2f a = {e0, e1}; v2f bw = {DXY, DXY};
        acc0 = wmma_acc(a, bw, acc0);
      }
      {
        const int i = i0 + 4 + kb;
        const float e0 = fminf(ex2(u[i]     + vm - Csh[i * PITCH + c0 + m]),       HUGEV);
        const float e1 = fminf(ex2(u[i + 1] + vm - Csh[(i + 1) * PITCH + c0 + m]), HUGEV);
        v2f a = {e0, e1}; v2f bw = {DXY, DXY};
        acc1 = wmma_acc(a, bw, acc1);
      }
    }
    if (m == 0) {
      const int cb = c0 + ((lane >> 4) << 3);
      #pragma unroll
      for (int r = 0; r < 8; ++r) scol[cb + r] = acc0[r] + acc1[r];  // col_marg_j
    }
  }
  __syncthreads();

  // ---- KL penalties + reductions
  {
    const float dv1 = srow[tid] / (1.0f + SMALLV);
    const float k1  = dv1 * __logf(dv1 + SMALLV) - dv1 + 1.0f;
    const float dv2 = scol[tid] / (1.0f + SMALLV);
    const float k2  = dv2 * __logf(dv2 + SMALLV) - dv2 + 1.0f;
    unew[tid] = (k1 + k2) * DXY;  // cons1 + cons2 contribution
  }
  #pragma unroll
  for (int off = 16; off > 0; off >>= 1) tacc += __shfl_down(tacc, off, 32);
  if (lane == 0) tred[wave] = tacc;
  __syncthreads();
  if (tid == 0) {
    float cons = 0.0f;
    for (int k = 0; k < IJ; ++k) cons += unew[k];
    float tr = 0.0f;
    for (int k = 0; k < 8; ++k) tr += tred[k];
    out[bat] = cons + tr * (1.0f / SSCALE) * DXY * DXY;  // unscale C' -> C
  }
}

extern "C" void kernel_launch(void* const* d_in, const int* in_sizes, int n_in,
                              void* d_out, int out_size, void* d_ws, size_t ws_size,
                              hipStream_t stream) {
  (void)in_sizes; (void)n_in; (void)d_ws; (void)ws_size; (void)out_size;
  const float* D = (const float*)d_in[0];
  float* out = (float*)d_out;
  const size_t shmem = (size_t)IJ * PITCH * sizeof(float);  // 260 KB of the 320 KB WGP LDS
  hipFuncSetAttribute(reinterpret_cast<const void*>(wfr_sinkhorn_kernel),
                      hipFuncAttributeMaxDynamicSharedMemorySize, (int)shmem);
  wfr_sinkhorn_kernel<<<dim3(256), dim3(256), shmem, stream>>>(D, out);
}